// GNN_14491219657351
// MI455X (gfx1250) — compile-verified
//
#include <hip/hip_runtime.h>

#define N_NODES 50000
#define N_EDGES 800000
#define F_IN    64
#define F_HID   128
#define F_OUT   64

typedef __attribute__((ext_vector_type(2))) float v2f;
typedef __attribute__((ext_vector_type(8))) float v8f;

// ---------------------------------------------------------------- zero ws
__global__ void k_zero(float* __restrict__ p, int n) {
    int i = blockIdx.x * blockDim.x + threadIdx.x;
    if (i < n) p[i] = 0.0f;
}

// ---------------------------------------------------------------- degree
__global__ void k_degree(const long long* __restrict__ edges,
                         float* __restrict__ cnt) {
    int e = blockIdx.x * blockDim.x + threadIdx.x;   // exact grid: E threads
    long long d = edges[N_EDGES + e];
    atomicAdd(&cnt[d], 1.0f);
}

// ------------------------------------------- scatter 64-wide features
// 16 threads per edge, 4 features per thread (float4 gather, 4 f32 atomics)
__global__ void k_scatter64(const long long* __restrict__ edges,
                            const float* __restrict__ x,
                            float* __restrict__ agg) {
    int t = blockIdx.x * blockDim.x + threadIdx.x;   // exact grid: E*16
    int e = t >> 4;
    int f = (t & 15) << 2;
    long long s = edges[e];
    long long d = edges[N_EDGES + e];
    const float4 v = *(const float4*)(x + s * 64 + f);
    float* a = agg + d * 64 + f;
    atomicAdd(a + 0, v.x);
    atomicAdd(a + 1, v.y);
    atomicAdd(a + 2, v.z);
    atomicAdd(a + 3, v.w);
}

// ------------------------------------------- scatter 128-wide features
__global__ void k_scatter128(const long long* __restrict__ edges,
                             const float* __restrict__ h,
                             float* __restrict__ agg) {
    int t = blockIdx.x * blockDim.x + threadIdx.x;   // exact grid: E*32
    int e = t >> 5;
    int f = (t & 31) << 2;
    long long s = edges[e];
    long long d = edges[N_EDGES + e];
    const float4 v = *(const float4*)(h + s * 128 + f);
    float* a = agg + d * 128 + f;
    atomicAdd(a + 0, v.x);
    atomicAdd(a + 1, v.y);
    atomicAdd(a + 2, v.z);
    atomicAdd(a + 3, v.w);
}

// ---------------------------------------------------------------- fused SAGE
// out[m,n] = act( (agg[m,:]/max(cnt[m],1)) @ Wl + x[m,:] @ Wr + bias[n] )
// One wave32 per 16x16 output tile. Uses exact-fp32 WMMA (16x16x4).
// A-frag (16x4 f32): lane L holds rows M=L%16, K pair = kbase + (L/16)*2.
// C/D (16x16 f32, 8 VGPRs): VGPR r -> M = r + 8*(L/16), N = L%16.
template<int K, int NOUT, bool RELU>
__global__ void k_sage(const float* __restrict__ agg,   // [N,K]
                       const float* __restrict__ cnt,   // [N]
                       const float* __restrict__ xin,   // [N,K]
                       const float* __restrict__ Wl,    // [K,NOUT] row-major
                       const float* __restrict__ Wr,    // [K,NOUT] row-major
                       const float* __restrict__ bias,  // [NOUT]
                       float* __restrict__ out)         // [N,NOUT]
{
    const int lane  = threadIdx.x & 31;
    const int wave  = threadIdx.x >> 5;            // output col tile
    const int m0    = blockIdx.x * 16;             // node tile base
    const int n     = wave * 16 + (lane & 15);     // output column
    const int khalf = (lane >> 4) * 2;             // 0 or 2
    const int mrow  = m0 + (lane & 15);            // A-matrix row

    const float inv = 1.0f / fmaxf(cnt[mrow], 1.0f);

    const float* __restrict__ arow = agg + (long)mrow * K;
    const float* __restrict__ xrow = xin + (long)mrow * K;

    v8f acc = {};
#pragma unroll
    for (int k = 0; k < K; k += 4) {
        const int ka = k + khalf;
        // mean-aggregated branch: scale A-frag by 1/deg on the fly
        v2f a1 = *(const v2f*)(arow + ka);
        a1 *= inv;
        v2f b1;
        b1.x = Wl[(long)ka * NOUT + n];
        b1.y = Wl[(long)(ka + 1) * NOUT + n];
        acc = __builtin_amdgcn_wmma_f32_16x16x4_f32(
                  false, a1, false, b1, (short)0, acc, false, false);
        // root branch
        v2f a2 = *(const v2f*)(xrow + ka);
        v2f b2;
        b2.x = Wr[(long)ka * NOUT + n];
        b2.y = Wr[(long)(ka + 1) * NOUT + n];
        acc = __builtin_amdgcn_wmma_f32_16x16x4_f32(
                  false, a2, false, b2, (short)0, acc, false, false);
    }

    const float bv    = bias[n];
    const int   mhalf = (lane >> 4) * 8;
#pragma unroll
    for (int r = 0; r < 8; r++) {
        float v = acc[r] + bv;
        if (RELU) v = fmaxf(v, 0.0f);
        out[(long)(m0 + mhalf + r) * NOUT + n] = v;
    }
}

// ---------------------------------------------------------------- launcher
extern "C" void kernel_launch(void* const* d_in, const int* in_sizes, int n_in,
                              void* d_out, int out_size, void* d_ws, size_t ws_size,
                              hipStream_t stream) {
    const float*     x    = (const float*)d_in[0];
    const long long* eidx = (const long long*)d_in[1];
    const float*     W1l  = (const float*)d_in[2];
    const float*     b1   = (const float*)d_in[3];
    const float*     W1r  = (const float*)d_in[4];
    const float*     W2l  = (const float*)d_in[5];
    const float*     b2   = (const float*)d_in[6];
    const float*     W2r  = (const float*)d_in[7];
    float*           out  = (float*)d_out;

    float* ws   = (float*)d_ws;
    float* cnt  = ws;                                  // [N]
    float* agg1 = cnt  + N_NODES;                      // [N,64]
    float* h    = agg1 + (size_t)N_NODES * F_IN;       // [N,128]
    float* agg2 = h    + (size_t)N_NODES * F_HID;      // [N,128]

    const int ws_floats = N_NODES * (1 + F_IN + F_HID + F_HID);
    k_zero<<<(ws_floats + 255) / 256, 256, 0, stream>>>(ws, ws_floats);

    k_degree<<<N_EDGES / 256, 256, 0, stream>>>(eidx, cnt);

    // Layer 1
    k_scatter64<<<(N_EDGES * 16) / 256, 256, 0, stream>>>(eidx, x, agg1);
    k_sage<F_IN, F_HID, true><<<N_NODES / 16, (F_HID / 16) * 32, 0, stream>>>(
        agg1, cnt, x, W1l, W1r, b1, h);

    // Layer 2
    k_scatter128<<<(N_EDGES * 32) / 256, 256, 0, stream>>>(eidx, h, agg2);
    k_sage<F_HID, F_OUT, false><<<N_NODES / 16, (F_OUT / 16) * 32, 0, stream>>>(
        agg2, cnt, h, W2l, W2r, b2, out);
}